// BaseValidationLoss_83210696393077
// MI455X (gfx1250) — compile-verified
//
#include <hip/hip_runtime.h>
#include <hip/hip_bf16.h>

#define Bc   4
#define Nc   1000000
#define Hc   128
#define Wc   128
#define HWc  (Hc * Wc)
#define WGPB 64          // workgroups per batch
#define TPB  256         // threads per block (8 wave32)
#define NBUF 4           // async pipeline depth

// ---------------------------------------------------------------------------
// Kernel 1: initialize output.
//   channels 0..1 (iwe)      <- 0           (splat kernel atomically adds)
//   channels 2..3 (avg_flow) <- flow * m / (m + 1e-9)   (matches JAX f32 math)
// ---------------------------------------------------------------------------
__global__ __launch_bounds__(256) void ev_init_kernel(
    const float* __restrict__ flow,        // [B,2,H,W]
    const float* __restrict__ event_mask,  // [B,1,H,W]
    float* __restrict__ out)               // [B,4,H,W]
{
    int i = blockIdx.x * blockDim.x + threadIdx.x;     // over B*2*HW
    if (i >= Bc * 2 * HWc) return;
    int b = i / (2 * HWc);
    int r = i - b * 2 * HWc;                           // c*HW + s
    int s = r & (HWc - 1);
    float m = event_mask[b * HWc + s];
    float* ob = out + (size_t)b * 4 * HWc;
    ob[r] = 0.0f;                                      // iwe channels
    ob[2 * HWc + r] = flow[i] * m / (m + 1e-9f);       // avg_flow channels
}

// ---------------------------------------------------------------------------
// Kernel 2: event warp + bilinear splat.
// Per block: private full 2-channel histogram in LDS (128 KB, possible only
// with CDNA5's 320 KB/WGP), events streamed with a depth-4 pipeline of
// CDNA5 global_load_async_to_lds_b128, corner accumulation via ds_add_f32.
// ---------------------------------------------------------------------------
__global__ __launch_bounds__(TPB) void ev_splat_kernel(
    const float* __restrict__ ev,    // [B,N,4] (t,y,x,p)
    const float* __restrict__ flow,  // [B,2,H,W]
    float* __restrict__ out)         // [B,4,H,W]
{
    extern __shared__ float smem[];
    float* hist  = smem;               // 2*HW floats  (pos @0, neg @HW)
    float* evbuf = smem + 2 * HWc;     // NBUF * TPB * 4 floats staging

    const int tid = threadIdx.x;
    const int b   = blockIdx.x / WGPB;
    const int wg  = blockIdx.x - b * WGPB;

    // vectorized zero of the private histogram (ds_store_b128)
    for (int i = tid * 4; i < 2 * HWc; i += TPB * 4)
        *(float4*)&hist[i] = make_float4(0.f, 0.f, 0.f, 0.f);
    __syncthreads();

    const int per = (Nc + WGPB - 1) / WGPB;            // 15625
    const int n0  = wg * per;
    const int n1  = min(Nc, n0 + per);
    const int nChunks = (n1 - n0 + TPB - 1) / TPB;

    const unsigned long long sbase = (unsigned long long)(ev + (size_t)b * Nc * 4);
    const float* fb = flow + (size_t)b * 2 * HWc;

    // raw LDS byte base addresses
    const unsigned ldsHist = (unsigned)(unsigned long long)(uintptr_t)&hist[0];
    const unsigned ldsEv   = (unsigned)(unsigned long long)(uintptr_t)&evbuf[0];

    // CDNA5 async copy: global (sbase + voff) -> LDS slot, 16 bytes.
    // Indices clamp to a valid event, so over-issue past the tail is safe.
#define ISSUE_CHUNK(ck)                                                       \
    {                                                                         \
        unsigned voff = (unsigned)min(n0 + (ck) * TPB + tid, Nc - 1) * 16u;   \
        unsigned la   = ldsEv + (unsigned)((((ck) & (NBUF - 1)) * TPB + tid) * 16); \
        asm volatile("global_load_async_to_lds_b128 %0, %1, %2"               \
                     :: "v"(la), "v"(voff), "s"(sbase) : "memory");           \
    }

    if (nChunks > 0) {                                 // prologue: 3 in flight
        ISSUE_CHUNK(0);
        ISSUE_CHUNK(1);
        ISSUE_CHUNK(2);
    }

    for (int c = 0; c < nChunks; ++c) {
        ISSUE_CHUNK(c + 3);                            // keep 4 in flight
        // issues so far = c+4; cnt<=3 => >= c+1 complete (in-order) => chunk c ready
        asm volatile("s_wait_asynccnt 0x3" ::: "memory");

        float4 e = *(const float4*)&evbuf[((c & (NBUF - 1)) * TPB + tid) * 4];

        const int cs = n0 + c * TPB;
        if (cs + tid < n1) {
            const float t = e.x, y = e.y, x = e.z, p = e.w;

            // gather per-event flow at rounded integer position (L2-resident)
            const int fi = __float2int_rn(y) * Wc + __float2int_rn(x);
            const float fx = fb[fi];          // flow channel 0 -> x
            const float fy = fb[HWc + fi];    // flow channel 1 -> y

            const float dt  = 1.0f - t;
            const float wyf = y + dt * fy;    // warped y
            const float wxf = x + dt * fx;    // warped x

            const float ty = floorf(wyf), lx = floorf(wxf);
            const float ay = wyf - ty;        // bottom-row weight
            const float ax = wxf - lx;        // right-col weight
            const int iy = (int)ty, ix = (int)lx;

            const unsigned hb = ldsHist + ((p > 0.5f) ? 0u : (unsigned)(HWc * 4));

            const bool y0 = (iy >= 0)     & (iy < Hc);
            const bool y1 = (iy + 1 >= 0) & (iy + 1 < Hc);
            const bool x0 = (ix >= 0)     & (ix < Wc);
            const bool x1 = (ix + 1 >= 0) & (ix + 1 < Wc);

            const float w00 = (1.0f - ay) * (1.0f - ax);
            const float w01 = (1.0f - ay) * ax;
            const float w10 = ay * (1.0f - ax);
            const float w11 = ay * ax;

            // guaranteed-native LDS f32 atomics (VDS ds_add_f32, RNE)
            if (y0 && x0 && w00 != 0.0f) {
                unsigned a = hb + (unsigned)((iy * Wc + ix) * 4);
                asm volatile("ds_add_f32 %0, %1" :: "v"(a), "v"(w00) : "memory");
            }
            if (y0 && x1 && w01 != 0.0f) {
                unsigned a = hb + (unsigned)((iy * Wc + ix + 1) * 4);
                asm volatile("ds_add_f32 %0, %1" :: "v"(a), "v"(w01) : "memory");
            }
            if (y1 && x0 && w10 != 0.0f) {
                unsigned a = hb + (unsigned)(((iy + 1) * Wc + ix) * 4);
                asm volatile("ds_add_f32 %0, %1" :: "v"(a), "v"(w10) : "memory");
            }
            if (y1 && x1 && w11 != 0.0f) {
                unsigned a = hb + (unsigned)(((iy + 1) * Wc + ix + 1) * 4);
                asm volatile("ds_add_f32 %0, %1" :: "v"(a), "v"(w11) : "memory");
            }
        }
    }
#undef ISSUE_CHUNK

    // inline-asm DS ops are not tracked by compiler bookkeeping: drain DScnt
    // so every wave's ds_add_f32 is visible before the cross-wave barrier.
    asm volatile("s_wait_dscnt 0x0" ::: "memory");
    __syncthreads();

    // flush private histogram into out[b, 0:2, :, :] (zero-skip)
    float* ob = out + (size_t)b * 4 * HWc;
    for (int i = tid * 4; i < 2 * HWc; i += TPB * 4) {
        float4 v = *(const float4*)&hist[i];
        if (v.x != 0.0f) unsafeAtomicAdd(&ob[i + 0], v.x);
        if (v.y != 0.0f) unsafeAtomicAdd(&ob[i + 1], v.y);
        if (v.z != 0.0f) unsafeAtomicAdd(&ob[i + 2], v.z);
        if (v.w != 0.0f) unsafeAtomicAdd(&ob[i + 3], v.w);
    }
}

extern "C" void kernel_launch(void* const* d_in, const int* in_sizes, int n_in,
                              void* d_out, int out_size, void* d_ws, size_t ws_size,
                              hipStream_t stream) {
    const float* flow       = (const float*)d_in[0];  // [B,2,H,W]
    const float* event_list = (const float*)d_in[1];  // [B,N,4]
    // d_in[2] = pol_mask is redundant (p is event_list[...,3]) -> not read
    const float* event_mask = (const float*)d_in[3];  // [B,1,H,W]
    float* out = (float*)d_out;                       // [B,4,H,W]

    // 1) zero iwe channels + write avg_flow channels
    {
        int tot = Bc * 2 * HWc;
        dim3 grid((tot + 255) / 256), block(256);
        hipLaunchKernelGGL(ev_init_kernel, grid, block, 0, stream,
                           flow, event_mask, out);
    }
    // 2) warp + splat (ordered after init on same stream)
    {
        size_t shmem = (size_t)(2 * HWc + NBUF * TPB * 4) * sizeof(float); // 144 KB
        dim3 grid(Bc * WGPB), block(TPB);
        hipLaunchKernelGGL(ev_splat_kernel, grid, block, shmem, stream,
                           event_list, flow, out);
    }
}